// SphereSampler_75505525064121
// MI455X (gfx1250) — compile-verified
//
#include <hip/hip_runtime.h>
#include <cstdint>

// Power-spherical forward: sample + entropy.
//   rows = 256*512 = 131072, each row = 256 f32 (255 mean dims + 1 raw conc)
//   out  = sample[rows*255] ++ entropy[rows]
// Mapping: one wave32 per row, 8 elements per lane. Row staged into LDS via
// CDNA5 async global->LDS b128 copies; all row reductions via shfl_xor.

#define PS_COUNT 512
#define PS_DIM   255
#define PS_BATCH 256
#define PS_ROWS  (PS_BATCH * PS_COUNT)   /* 131072 */
#define PS_EPS   1e-6f

// 4 x i32 vector type, as required by the async-LDS builtins
typedef int v4i_t __attribute__((vector_size(16)));
typedef v4i_t __attribute__((address_space(1)))* glb_v4i_ptr;   // global (AS1)
typedef v4i_t __attribute__((address_space(3)))* lds_v4i_ptr;   // LDS (AS3)

// ---------- deterministic counter-based RNG ----------
__device__ __forceinline__ uint32_t pcg(uint32_t v) {
  uint32_t s = v * 747796405u + 2891336453u;
  uint32_t w = ((s >> ((s >> 28u) + 4u)) ^ s) * 277803737u;
  return (w >> 22u) ^ w;
}
__device__ __forceinline__ float u01(uint32_t h) {
  return ((float)(h >> 8) + 0.5f) * (1.0f / 16777216.0f);   // (0,1)
}
__device__ __forceinline__ float rnorm(uint32_t tag) {
  float u1 = u01(pcg(tag * 2u + 0x68E31DA4u));
  float u2 = u01(pcg(tag * 2u + 0xB5297A4Du));
  return sqrtf(-2.0f * __logf(u1)) * __cosf(6.283185307179586f * u2);
}
// Marsaglia-Tsang gamma, fixed 4 unrolled attempts (branchless accept)
__device__ __forceinline__ float rgamma(float a, uint32_t tag) {
  float d = a - (1.0f / 3.0f);
  float c = rsqrtf(9.0f * d);
  float res = d;            // fallback ~ mean
  bool done = false;
#pragma unroll
  for (int i = 0; i < 4; ++i) {
    uint32_t t = tag + 0x3C6EF372u * (uint32_t)i;
    float xn = rnorm(t);
    float v1 = 1.0f + c * xn;
    float v3 = v1 * v1 * v1;
    float u  = u01(pcg(t ^ 0x1B873593u));
    bool acc = (v1 > 0.0f) &&
               (__logf(u) < 0.5f * xn * xn + d - d * v3 +
                            d * __logf(fmaxf(v3, 1e-30f)));
    if (acc && !done) { res = d * v3; done = true; }
  }
  return res;
}

// ---------- special functions (args >= 127 -> asymptotics exact in f32) ----------
__device__ __forceinline__ float lgamma_stirling(float x) {
  const float HALF_LN_2PI = 0.9189385332046727f;
  float inv = 1.0f / x, inv2 = inv * inv;
  return (x - 0.5f) * __logf(x) - x + HALF_LN_2PI +
         inv * (1.0f / 12.0f) - inv * inv2 * (1.0f / 360.0f);
}
__device__ __forceinline__ float digamma_asym(float x) {
  float inv = 1.0f / x, inv2 = inv * inv;
  return __logf(x) - 0.5f * inv -
         inv2 * (1.0f / 12.0f - inv2 * (1.0f / 120.0f - inv2 * (1.0f / 252.0f)));
}

// ---------- wave32 all-reduce ----------
__device__ __forceinline__ float wave_sum(float v) {
#pragma unroll
  for (int off = 16; off >= 1; off >>= 1) v += __shfl_xor(v, off, 32);
  return v;
}

__global__ __launch_bounds__(256) void
SphereSampler_75505525064121_kernel(const float* __restrict__ x,
                                    float* __restrict__ out_sample,
                                    float* __restrict__ out_entropy)
{
  __shared__ __align__(16) float lds[8 * 256];   // 8 rows per block (1 per wave)
  const int lane = threadIdx.x & 31;
  const int wid  = threadIdx.x >> 5;
  const int row  = (int)blockIdx.x * 8 + wid;

  float* rowLds = lds + wid * 256;
  const float* gsrc = x + (size_t)row * 256 + lane * 8;   // 32B per lane, 1KB per wave

#if __has_builtin(__builtin_amdgcn_global_load_async_to_lds_b128) && \
    __has_builtin(__builtin_amdgcn_s_wait_asynccnt)
  {
    // CDNA5 async global->LDS DMA (CDNA5 ISA), ASYNCcnt tracked
    glb_v4i_ptr g = (glb_v4i_ptr)(uintptr_t)(const void*)gsrc;
    lds_v4i_ptr l = (lds_v4i_ptr)(uint32_t)(uintptr_t)(void*)(rowLds + lane * 8);
    __builtin_amdgcn_global_load_async_to_lds_b128(g, l, 0, 0);
    __builtin_amdgcn_global_load_async_to_lds_b128(g, l, 16, 0);
    __builtin_amdgcn_s_wait_asynccnt(0);
  }
#else
  {
    const float4* g4 = (const float4*)gsrc;
    float4* l4 = (float4*)(rowLds + lane * 8);
    l4[0] = g4[0];
    l4[1] = g4[1];
  }
#endif
  __syncthreads();

  float m[8];
  {
    const float4 a = ((const float4*)(rowLds + lane * 8))[0];
    const float4 b = ((const float4*)(rowLds + lane * 8))[1];
    m[0] = a.x; m[1] = a.y; m[2] = a.z; m[3] = a.w;
    m[4] = b.x; m[5] = b.y; m[6] = b.z; m[7] = b.w;
  }
  const int e0 = lane * 8;

  // ---- normalize mean (elements 0..254) ----
  float s2p = 0.0f;
#pragma unroll
  for (int j = 0; j < 8; ++j) { int e = e0 + j; if (e < PS_DIM) s2p += m[j] * m[j]; }
  const float rs = rsqrtf(wave_sum(s2p));

  // ---- concentration from raw element 255 (LDS broadcast read) ----
  const float xl   = rowLds[255];
  const float sp   = fmaxf(xl, 0.0f) + log1pf(__expf(-fabsf(xl)));  // softplus
  const float conc = 1.0f + PS_EPS + sp;
  const float alpha = 127.0f + conc;      // half = (DIM-1)/2 = 127
  const float apb   = alpha + 127.0f;

  // ---- z ~ Beta(alpha, 127) via two gammas; identical on all lanes ----
  const uint32_t baser = (uint32_t)row * 0x9E3779B9u;
  const float ga = rgamma(alpha,  baser ^ 0xA511E9B3u);
  const float gb = rgamma(127.0f, baser ^ 0x63D83595u);
  const float z  = ga / (ga + gb);
  const float t  = 2.0f * z - 1.0f;

  // ---- tangent direction v in R^254 (elements 1..254) ----
  float n[8];
#pragma unroll
  for (int j = 0; j < 8; ++j) {
    int e = e0 + j;
    n[j] = rnorm(baser + 0x85EBCA6Bu * (uint32_t)(e + 1));
  }
  float svp = 0.0f;
#pragma unroll
  for (int j = 0; j < 8; ++j) { int e = e0 + j; if (e >= 1 && e < PS_DIM) svp += n[j] * n[j]; }
  const float rv    = rsqrtf(wave_sum(svp));
  const float scale = sqrtf(fmaxf(1.0f - t * t, PS_EPS)) * rv;

  // ---- y, Householder direction u = normalize(e1 - mean) ----
  float y[8], uh[8];
  float nup = 0.0f;
#pragma unroll
  for (int j = 0; j < 8; ++j) {
    int e = e0 + j;
    float mn = m[j] * rs;
    float uv = (e == 0) ? (1.0f - mn) : -mn;
    float yv = (e == 0) ? t : n[j] * scale;
    if (e >= PS_DIM) { uv = 0.0f; yv = 0.0f; }
    uh[j] = uv; nup += uv * uv;
    y[j]  = yv;
  }
  const float invnu = 1.0f / fmaxf(sqrtf(wave_sum(nup)), PS_EPS);
  float dotp = 0.0f;
#pragma unroll
  for (int j = 0; j < 8; ++j) { uh[j] *= invnu; dotp += y[j] * uh[j]; }
  const float dot2 = 2.0f * wave_sum(dotp);

  // ---- sample = y - 2 (y.u) u ----
  float* orow = out_sample + (size_t)row * PS_DIM;
#pragma unroll
  for (int j = 0; j < 8; ++j) {
    int e = e0 + j;
    if (e < PS_DIM) orow[e] = y[j] - dot2 * uh[j];
  }

  // ---- entropy (one value per row) ----
  if (lane == 0) {
    const float LN2  = 0.6931471805599453f;
    const float LNPI = 1.1447298858494002f;
    float log_norm = -(apb * LN2 + lgamma_stirling(alpha) - lgamma_stirling(apb) +
                       127.0f * LNPI);
    out_entropy[row] =
        -(log_norm + conc * (LN2 + digamma_asym(alpha) - digamma_asym(apb)));
  }
}

extern "C" void kernel_launch(void* const* d_in, const int* in_sizes, int n_in,
                              void* d_out, int out_size, void* d_ws, size_t ws_size,
                              hipStream_t stream) {
  (void)in_sizes; (void)n_in; (void)out_size; (void)d_ws; (void)ws_size;
  const float* x = (const float*)d_in[0];
  float* out = (float*)d_out;
  float* out_sample  = out;
  float* out_entropy = out + (size_t)PS_ROWS * PS_DIM;
  hipLaunchKernelGGL(SphereSampler_75505525064121_kernel,
                     dim3(PS_ROWS / 8), dim3(256), 0, stream,
                     x, out_sample, out_entropy);
}